// GraphInvariantPointAttention_87935160418345
// MI455X (gfx1250) — compile-verified
//
#include <hip/hip_runtime.h>
#include <hip/hip_bf16.h>
#include <math.h>

// ---------------------------------------------------------------------------
// Graph Invariant Point Attention for MI455X (gfx1250, wave32).
// Dense GEMMs: V_WMMA_F32_16X16X4_F32, one wave per 16x(16*NT) tile,
// branch-free inner loop. A/B streams addressed as SGPR-base + 32-bit
// VGPR offsets so the backend emits GLOBAL (GVS) loads, not FLAT.
// Edge gather/scatter phase is HBM-bound (~3 GB of gathers @ 23.3 TB/s).
// ---------------------------------------------------------------------------

typedef float v2f __attribute__((ext_vector_type(2)));
typedef float v8f __attribute__((ext_vector_type(8)));

static constexpr int N_NODES = 20000;
static constexpr int N_EDGES = 640000;
static constexpr int CS = 384;
static constexpr int CZ = 128;
static constexpr int CH = 16;
static constexpr int NH = 12;     // heads
static constexpr int PQ = 4;
static constexpr int PV = 8;

// ---------------------------------------------------------------------------
// fp32 WMMA GEMM:  C[M x Nout] = A[M x K] * W[K x Nout] + bias
// One wave computes a 16 x (16*NT) tile: one A fragment drives NT WMMAs.
// Out-of-range columns are handled by clamping the offset (their D columns
// are garbage but never stored) -> no divergence in the K loop, EXEC stays
// all-ones for every WMMA as the ISA requires.
// All offsets are 32-bit elements (largest buffer here is ~327 MB < 2^31 B).
// M, K must be multiples of 16 / 4 (true at all call sites).
// ---------------------------------------------------------------------------
template <int NT>
__global__ __launch_bounds__(32)
void wmma_gemm_f32(const float* __restrict__ A, const float* __restrict__ W,
                   const float* __restrict__ bias, float* __restrict__ C,
                   int M, int K, int Nout) {
  const int lane = threadIdx.x & 31;
  const int half = lane >> 4;        // 0: lanes 0-15, 1: lanes 16-31
  const int l15  = lane & 15;
  const int tm = blockIdx.x;
  const int tn = blockIdx.y;
  const int m  = tm * 16 + l15;      // A row held by this lane
  const int kp = half * 2;           // K sub-offset for A/B fragments

  // Per-subtile B offsets (32-bit, strength-reduced: += 4*Nout per K step).
  unsigned woff[NT];
#pragma unroll
  for (int s = 0; s < NT; ++s) {
    int n = tn * (16 * NT) + s * 16 + l15;
    unsigned nc = (n < Nout) ? (unsigned)n : 0u;   // clamp: garbage col, never stored
    woff[s] = (unsigned)(kp * Nout) + nc;
  }

  v8f acc[NT];
#pragma unroll
  for (int s = 0; s < NT; ++s) acc[s] = (v8f){0.f,0.f,0.f,0.f,0.f,0.f,0.f,0.f};

  unsigned aoff = (unsigned)m * (unsigned)K + (unsigned)kp;
  const unsigned wstep = (unsigned)(4 * Nout);
  const unsigned nstride = (unsigned)Nout;

  for (int k = 0; k < K; k += 4) {
    v2f av = *(const v2f*)(A + aoff);      // A[m][k+kp], A[m][k+kp+1]
    aoff += 4;
#pragma unroll
    for (int s = 0; s < NT; ++s) {
      v2f bv;
      bv.x = W[woff[s]];                   // W[k+kp  ][n]
      bv.y = W[woff[s] + nstride];         // W[k+kp+1][n]
      woff[s] += wstep;
      acc[s] = __builtin_amdgcn_wmma_f32_16x16x4_f32(
          false, av, false, bv, (short)0, acc[s], false, false);
    }
  }

#pragma unroll
  for (int s = 0; s < NT; ++s) {
    const int n = tn * (16 * NT) + s * 16 + l15;
    if (n < Nout) {
      const float b = bias[n];
#pragma unroll
      for (int r = 0; r < 8; ++r) {
        const unsigned row = (unsigned)(tm * 16 + r + half * 8);   // C/D layout
        C[row * nstride + (unsigned)n] = acc[s][r] + b;
      }
    }
  }
}

// ---------------------------------------------------------------------------
// Apply per-node rigid transform to raw point projections.
// raw row layout (from s@W): [d=0 block | d=1 block | d=2 block], each of
// size npts_total = H*npts  (this is the jnp.split/stack semantics).
// out row layout: (npts_total, 3) interleaved xyz.
// ---------------------------------------------------------------------------
__global__ __launch_bounds__(256)
void transform_points(const float* __restrict__ raw, const float* __restrict__ rot,
                      const float* __restrict__ trans, float* __restrict__ out,
                      int npts_total, int N) {
  int n = blockIdx.x * blockDim.x + threadIdx.x;
  if (n >= N) return;
  const float* R = rot + (size_t)n * 9;         // row-major 3x3
  const float r00=R[0],r01=R[1],r02=R[2],r10=R[3],r11=R[4],r12=R[5],
              r20=R[6],r21=R[7],r22=R[8];
  const float tx=trans[n*3+0], ty=trans[n*3+1], tz=trans[n*3+2];
  const float* row = raw + (size_t)n * (npts_total * 3);
  float* orow = out + (size_t)n * (npts_total * 3);
  for (int i = 0; i < npts_total; ++i) {
    const float x = row[i], y = row[npts_total + i], z = row[2*npts_total + i];
    orow[i*3+0] = r00*x + r01*y + r02*z + tx;
    orow[i*3+1] = r10*x + r11*y + r12*z + ty;
    orow[i*3+2] = r20*x + r21*y + r22*z + tz;
  }
}

// ---------------------------------------------------------------------------
// Per-edge attention logits.
// ---------------------------------------------------------------------------
__global__ __launch_bounds__(256)
void edge_logits(const float* __restrict__ q, const float* __restrict__ kv,
                 const float* __restrict__ q_pts, const float* __restrict__ kv_pts,
                 const float* __restrict__ b_edge, const float* __restrict__ head_w,
                 const int* __restrict__ ei, float* __restrict__ a, int E) {
  int e = blockIdx.x * blockDim.x + threadIdx.x;
  if (e >= E) return;
  const int i0 = ei[e];
  const int i1 = ei[E + e];
  const float inv_sqrt48 = 0.14433756729740643f;   // sqrt(1/(3*CH))
  const float inv_sqrt3  = 0.57735026918962576f;   // sqrt(1/3)
  const float wcoef      = 0.13608276348795434f;   // sqrt(1/(3*PQ*9/2))
  const float* qrow = q      + (size_t)i1 * (NH*CH);
  const float* krow = kv     + (size_t)i0 * (NH*2*CH);
  const float* qp   = q_pts  + (size_t)i1 * (NH*PQ*3);
  const float* kp   = kv_pts + (size_t)i0 * (NH*(PQ+PV)*3);
  for (int h = 0; h < NH; ++h) {
    float qk = 0.f;
#pragma unroll
    for (int c = 0; c < CH; ++c) qk += qrow[h*CH + c] * krow[h*2*CH + c];
    float pt = 0.f;
#pragma unroll
    for (int p = 0; p < PQ; ++p) {
      const float dx = qp[h*PQ*3 + p*3 + 0] - kp[h*(PQ+PV)*3 + p*3 + 0];
      const float dy = qp[h*PQ*3 + p*3 + 1] - kp[h*(PQ+PV)*3 + p*3 + 1];
      const float dz = qp[h*PQ*3 + p*3 + 2] - kp[h*(PQ+PV)*3 + p*3 + 2];
      pt += dx*dx + dy*dy + dz*dz;
    }
    const float hw = log1pf(__expf(head_w[h])) * wcoef;   // softplus * scale
    a[(size_t)e*NH + h] = qk * inv_sqrt48 + inv_sqrt3 * b_edge[(size_t)e*NH + h]
                          - 0.5f * hw * pt;
  }
}

// Order-preserving float<->uint encoding for atomicMax-based segment max.
__device__ __forceinline__ unsigned enc_f(float f) {
  unsigned u = __float_as_uint(f);
  return (u & 0x80000000u) ? ~u : (u | 0x80000000u);
}
__device__ __forceinline__ float dec_f(unsigned u) {
  return (u & 0x80000000u) ? __uint_as_float(u & 0x7FFFFFFFu)
                           : __uint_as_float(~u);
}

__global__ __launch_bounds__(256)
void seg_max(const float* __restrict__ a, const int* __restrict__ ei,
             unsigned* __restrict__ mxb, int E) {
  int t = blockIdx.x * blockDim.x + threadIdx.x;
  if (t >= E * NH) return;
  const int e = t / NH, h = t % NH;
  const int seg = ei[E + e];                     // segment by ei1
  atomicMax(&mxb[(size_t)seg * NH + h], enc_f(a[t]));
}

__global__ __launch_bounds__(256)
void seg_exp_sum(float* __restrict__ a, const int* __restrict__ ei,
                 const unsigned* __restrict__ mxb, float* __restrict__ den, int E) {
  int t = blockIdx.x * blockDim.x + threadIdx.x;
  if (t >= E * NH) return;
  const int e = t / NH, h = t % NH;
  const int seg = ei[E + e];
  const float ex = __expf(a[t] - dec_f(mxb[(size_t)seg * NH + h]));
  a[t] = ex;
  atomicAdd(&den[(size_t)seg * NH + h], ex);
}

// ---------------------------------------------------------------------------
// Weighted scatter: o / o_pair accumulate straight into feats slices,
// o_pt into a separate (N, H, PV, 3) accumulator.
// feats row (960): [o 0..191 | x 192..287 | y 288..383 | z 384..479 |
//                   norm 480..575 | o_pair 576..959]
// ---------------------------------------------------------------------------
__global__ __launch_bounds__(256)
void aggregate(const float* __restrict__ a, const float* __restrict__ den,
               const float* __restrict__ kv, const float* __restrict__ kv_pts,
               const float* __restrict__ pair_z, const int* __restrict__ ei,
               float* __restrict__ feats, float* __restrict__ opt, int E) {
  int t = blockIdx.x * blockDim.x + threadIdx.x;
  if (t >= E * NH) return;
  const int e = t / NH, h = t % NH;
  const int i0 = ei[e];
  const int i1 = ei[E + e];
  const float w = a[t] / (den[(size_t)i1 * NH + h] + 1e-16f);

  float* fo = feats + (size_t)i0 * 960;
  const float* vrow = kv + (size_t)i1 * (NH*2*CH) + h*2*CH + CH;   // v half
#pragma unroll
  for (int c = 0; c < CH; ++c) atomicAdd(&fo[h*CH + c], w * vrow[c]);

  const float* vp = kv_pts + (size_t)i1 * (NH*(PQ+PV)*3) + h*(PQ+PV)*3 + PQ*3;
  float* op = opt + (size_t)i0 * (NH*PV*3) + h*PV*3;
#pragma unroll
  for (int j = 0; j < PV*3; ++j) atomicAdd(&op[j], w * vp[j]);

  const float* pz = pair_z + (size_t)e * 32;
#pragma unroll
  for (int c = 0; c < 32; ++c) atomicAdd(&fo[576 + h*32 + c], w * pz[c]);
}

// ---------------------------------------------------------------------------
// Finalize: inverse-rotate o_pt, write xyz + norm slices of feats.
// ---------------------------------------------------------------------------
__global__ __launch_bounds__(256)
void finalize_feats(const float* __restrict__ opt, const float* __restrict__ rot,
                    const float* __restrict__ trans, float* __restrict__ feats,
                    int N) {
  int t = blockIdx.x * blockDim.x + threadIdx.x;
  if (t >= N * NH) return;
  const int n = t / NH, h = t % NH;
  const float* R = rot + (size_t)n * 9;
  const float tx=trans[n*3+0], ty=trans[n*3+1], tz=trans[n*3+2];
  const float* op = opt + (size_t)n * (NH*PV*3) + h*PV*3;
  float* f = feats + (size_t)n * 960;
#pragma unroll
  for (int p = 0; p < PV; ++p) {
    const float dx = op[p*3+0] - tx;
    const float dy = op[p*3+1] - ty;
    const float dz = op[p*3+2] - tz;
    // einsum('nji,nhpj->nhpi'): out_i = sum_j R[j][i] * d_j  (R^T)
    const float rx = R[0]*dx + R[3]*dy + R[6]*dz;
    const float ry = R[1]*dx + R[4]*dy + R[7]*dz;
    const float rz = R[2]*dx + R[5]*dy + R[8]*dz;
    const int i = h*PV + p;
    f[192 + i] = rx;
    f[288 + i] = ry;
    f[384 + i] = rz;
    f[480 + i] = sqrtf(rx*rx + ry*ry + rz*rz + 1e-8f);
  }
}

// ---------------------------------------------------------------------------
extern "C" void kernel_launch(void* const* d_in, const int* in_sizes, int n_in,
                              void* d_out, int out_size, void* d_ws, size_t ws_size,
                              hipStream_t stream) {
  (void)in_sizes; (void)n_in; (void)out_size; (void)ws_size;
  const float* s       = (const float*)d_in[0];
  const float* z       = (const float*)d_in[1];
  const int*   ei      = (const int*)  d_in[2];   // (2, E) flat
  const float* rot     = (const float*)d_in[3];
  const float* trans   = (const float*)d_in[4];
  const float* W_q     = (const float*)d_in[5];
  const float* b_q     = (const float*)d_in[6];
  const float* W_kv    = (const float*)d_in[7];
  const float* b_kv    = (const float*)d_in[8];
  const float* W_qp    = (const float*)d_in[9];
  const float* b_qp    = (const float*)d_in[10];
  const float* W_kvp   = (const float*)d_in[11];
  const float* b_kvp   = (const float*)d_in[12];
  const float* W_b     = (const float*)d_in[13];
  const float* b_b     = (const float*)d_in[14];
  const float* W_down  = (const float*)d_in[15];
  const float* b_down  = (const float*)d_in[16];
  const float* head_w  = (const float*)d_in[17];
  const float* W_out   = (const float*)d_in[18];
  const float* b_out   = (const float*)d_in[19];
  float* out = (float*)d_out;

  // ---- workspace layout (floats) ----
  float* w = (float*)d_ws;
  size_t off = 0;
  float*    q      = w + off; off += (size_t)N_NODES * 192;            // 3.84M
  float*    kv     = w + off; off += (size_t)N_NODES * 384;            // 7.68M
  float*    q_pts  = w + off; off += (size_t)N_NODES * 144;            // 2.88M
  float*    kv_pts = w + off; off += (size_t)N_NODES * 432;            // 8.64M
  float*    b_edge = w + off; off += (size_t)N_EDGES * NH;             // 7.68M
  float*    pair_z = w + off; off += (size_t)N_EDGES * 32;             // 20.48M
  // --- zeroed-every-call region (contiguous) ---
  size_t zero_begin = off;
  unsigned* mxb    = (unsigned*)(w + off); off += (size_t)N_NODES * NH;
  float*    den    = w + off; off += (size_t)N_NODES * NH;
  float*    opt    = w + off; off += (size_t)N_NODES * 288;            // o_pt acc
  float*    feats  = w + off; off += (size_t)N_NODES * 960;
  size_t zero_count = off - zero_begin;
  // --- scratch reused across phases: raw point proj (8.64M) then logits ---
  float*    raw    = w + off; off += (size_t)N_NODES * 432;
  float*    a      = raw;                      // E*NH = 7.68M <= 8.64M

  hipMemsetAsync(w + zero_begin, 0, zero_count * sizeof(float), stream);

  const dim3 wblk(32);
  // Node projections (M = 20000 rows -> 1250 row tiles, 64-wide col tiles).
  wmma_gemm_f32<4><<<dim3(N_NODES/16, 3), wblk, 0, stream>>>(s, W_q,  b_q,  q,  N_NODES, CS, 192);
  wmma_gemm_f32<4><<<dim3(N_NODES/16, 6), wblk, 0, stream>>>(s, W_kv, b_kv, kv, N_NODES, CS, 384);
  wmma_gemm_f32<4><<<dim3(N_NODES/16, 3), wblk, 0, stream>>>(s, W_qp, b_qp, raw, N_NODES, CS, 144);
  transform_points<<<(N_NODES+255)/256, 256, 0, stream>>>(raw, rot, trans, q_pts, NH*PQ, N_NODES);
  wmma_gemm_f32<4><<<dim3(N_NODES/16, 7), wblk, 0, stream>>>(s, W_kvp, b_kvp, raw, N_NODES, CS, 432);
  transform_points<<<(N_NODES+255)/256, 256, 0, stream>>>(raw, rot, trans, kv_pts, NH*(PQ+PV), N_NODES);

  // Edge projections (M = 640000 rows -> 40000 row tiles, narrow outputs).
  wmma_gemm_f32<1><<<dim3(N_EDGES/16, 1), wblk, 0, stream>>>(z, W_b,    b_b,    b_edge, N_EDGES, CZ, NH);
  wmma_gemm_f32<2><<<dim3(N_EDGES/16, 1), wblk, 0, stream>>>(z, W_down, b_down, pair_z, N_EDGES, CZ, 32);

  // Edge attention. (`a` aliases `raw`, which is dead after point transforms.)
  edge_logits<<<(N_EDGES+255)/256, 256, 0, stream>>>(q, kv, q_pts, kv_pts, b_edge, head_w, ei, a, N_EDGES);

  const int EH = N_EDGES * NH;
  seg_max    <<<(EH+255)/256, 256, 0, stream>>>(a, ei, mxb, N_EDGES);
  seg_exp_sum<<<(EH+255)/256, 256, 0, stream>>>(a, ei, mxb, den, N_EDGES);
  aggregate  <<<(EH+255)/256, 256, 0, stream>>>(a, den, kv, kv_pts, pair_z, ei, feats, opt, N_EDGES);

  finalize_feats<<<(N_NODES*NH+255)/256, 256, 0, stream>>>(opt, rot, trans, feats, N_NODES);

  // Output GEMM: feats (20000 x 960) @ W_out (960 x 384) + b_out.
  wmma_gemm_f32<4><<<dim3(N_NODES/16, 6), wblk, 0, stream>>>(feats, W_out, b_out, out, N_NODES, 960, 384);
}